// VectorQuantize_1726576854453
// MI455X (gfx1250) — compile-verified
//
#include <hip/hip_runtime.h>
#include <hip/hip_bf16.h>
#include <math.h>

// Problem constants (from reference setup_inputs)
#define BATCH 16
#define DIM   1024
#define TLEN  4096
#define CS    1024
#define CD    8
#define TPB   256            // threads per block == timesteps per block in k_main/k_out

typedef __attribute__((ext_vector_type(2))) float v2f;
typedef __attribute__((ext_vector_type(8))) float v8f;

// ---------------------------------------------------------------------------
// Prep 1: W_in^T[d][o] = g_in[o] * V_in[o][d] / ||V_in[o]||   (weight_norm dim=0)
// ---------------------------------------------------------------------------
__global__ void __launch_bounds__(256) k_prep_win(const float* __restrict__ V_in,
                                                  const float* __restrict__ g_in,
                                                  float* __restrict__ Wt) {
  __shared__ float red[256];
  __shared__ float scale[CD];
  const int tid = threadIdx.x;
  for (int o = 0; o < CD; ++o) {
    float s = 0.f;
    for (int d = tid; d < DIM; d += 256) {
      float v = V_in[o * DIM + d];
      s = fmaf(v, v, s);
    }
    red[tid] = s;
    __syncthreads();
    for (int w = 128; w > 0; w >>= 1) {
      if (tid < w) red[tid] += red[tid + w];
      __syncthreads();
    }
    if (tid == 0) scale[o] = g_in[o] / sqrtf(red[0]);
    __syncthreads();
  }
  for (int o = 0; o < CD; ++o) {
    const float sc = scale[o];
    for (int d = tid; d < DIM; d += 256) Wt[d * CD + o] = V_in[o * DIM + d] * sc;
  }
}

// ---------------------------------------------------------------------------
// Prep 2: W_out rows (weight_norm) and codebook rows normalized + packed into
// WMMA-A layout.  A-tile (16x4 f32): value(lane,v) = A[lane&15][(lane>>4)*2+v].
// cbA[((tile*2+chunk)*32 + lane)*2 + v] = cb_n[tile*16+(lane&15)][chunk*4+(lane>>4)*2+v]
// ---------------------------------------------------------------------------
__global__ void __launch_bounds__(256) k_prep_rows(const float* __restrict__ V_out,
                                                   const float* __restrict__ g_out,
                                                   const float* __restrict__ codebook,
                                                   float* __restrict__ Wo,
                                                   float* __restrict__ cbA) {
  const int i = blockIdx.x * 256 + threadIdx.x;
  if (i < DIM) {
    float v[CD];
    float s = 0.f;
#pragma unroll
    for (int k = 0; k < CD; ++k) {
      v[k] = V_out[i * CD + k];
      s = fmaf(v[k], v[k], s);
    }
    const float sc = g_out[i] / sqrtf(s);
#pragma unroll
    for (int k = 0; k < CD; ++k) Wo[i * CD + k] = v[k] * sc;
  } else if (i < DIM + CS) {
    const int c = i - DIM;
    float v[CD];
    float s = 0.f;
#pragma unroll
    for (int k = 0; k < CD; ++k) {
      v[k] = codebook[c * CD + k];
      s = fmaf(v[k], v[k], s);
    }
    const float inv = 1.0f / fmaxf(sqrtf(s), 1e-12f);  // F.normalize eps
    const int tile = c >> 4;
    const int m    = c & 15;
#pragma unroll
    for (int chunk = 0; chunk < 2; ++chunk)
#pragma unroll
      for (int hi = 0; hi < 2; ++hi)
#pragma unroll
        for (int vv = 0; vv < 2; ++vv) {
          const int k = chunk * 4 + hi * 2 + vv;
          cbA[(((tile * 2 + chunk) * 32) + (m + (hi << 4))) * 2 + vv] = v[k] * inv;
        }
  }
}

// ---------------------------------------------------------------------------
// Main fused kernel: in-proj -> z_e, L2-normalize, WMMA codebook argmax,
// z_q lookup, per-block loss partial.  One block = one batch x 256 timesteps.
// ---------------------------------------------------------------------------
__global__ void __launch_bounds__(256) k_main(const float* __restrict__ z,
                                              const float* __restrict__ b_in,
                                              const float* __restrict__ codebook,
                                              const float* __restrict__ Wt,
                                              const float* __restrict__ cbA,
                                              float* __restrict__ o_idx,
                                              float* __restrict__ o_ze,
                                              float* __restrict__ partial) {
  __shared__ float smem[16384];              // 64 KB
  float* sW   = smem;                        // [1024*8]   phase 1 only
  float* sCbA = smem + 8192;                 // [8192]     whole kernel
  float* sEnc = smem;                        // [256*8]    reuses sW after phase 1
  int*   sIdx = (int*)(smem + 2048);         // [256]
  float* sRed = smem + 4096;                 // [256]

  const int tid = threadIdx.x;
  const int blk = blockIdx.x;
  const int b   = blk >> 4;                  // 16 blocks per batch (T/256)
  const int tt0 = (blk & 15) << 8;
  const int t   = tt0 + tid;

  for (int i = tid; i < 8192; i += 256) {
    sW[i]   = Wt[i];
    sCbA[i] = cbA[i];
  }
  __syncthreads();

  // ---- Phase 1: z_e[:,t] = W_in @ z[b,:,t] + b_in (8 accumulators/thread) ----
  float acc[CD];
#pragma unroll
  for (int o = 0; o < CD; ++o) acc[o] = b_in[o];

  const float* zp = z + ((size_t)b << 22) + t;   // b*DIM*TLEN
#pragma unroll 4
  for (int d = 0; d < DIM; ++d) {
    const float x = zp[(size_t)d << 12];
    const float4* w4 = (const float4*)(sW + d * CD);
    const float4 w0 = w4[0], w1 = w4[1];
    acc[0] = fmaf(w0.x, x, acc[0]);
    acc[1] = fmaf(w0.y, x, acc[1]);
    acc[2] = fmaf(w0.z, x, acc[2]);
    acc[3] = fmaf(w0.w, x, acc[3]);
    acc[4] = fmaf(w1.x, x, acc[4]);
    acc[5] = fmaf(w1.y, x, acc[5]);
    acc[6] = fmaf(w1.z, x, acc[6]);
    acc[7] = fmaf(w1.w, x, acc[7]);
  }

  // z_e output [B, CD, T]
#pragma unroll
  for (int o = 0; o < CD; ++o)
    o_ze[(((size_t)b * CD + o) << 12) + t] = acc[o];

  // ---- Phase 2: L2 normalize encoding ----
  float ss = 0.f;
#pragma unroll
  for (int o = 0; o < CD; ++o) ss = fmaf(acc[o], acc[o], ss);
  const float inv = 1.0f / fmaxf(sqrtf(ss), 1e-12f);

  __syncthreads();                                   // all waves done reading sW
#pragma unroll
  for (int o = 0; o < CD; ++o) sEnc[tid * CD + o] = acc[o] * inv;
  __syncthreads();

  // ---- Phase 3: scores = cb_n(16x8) @ enc_n^T(8x16) via V_WMMA_F32_16X16X4_F32,
  //      argmax over 1024 codes.  D layout: lane l, vgpr v -> code (l>>4)*8+v,
  //      timestep l&15  => per-lane argmax, one shfl_xor(16) merge at the end. ----
  const int lane = tid & 31;
  const int wave = tid >> 5;
  const int nn   = lane & 15;
  const int hi   = lane >> 4;

  for (int g = 0; g < 2; ++g) {
    const int tl = wave * 32 + g * 16 + nn;          // timestep this column serves
    const float* e = sEnc + tl * CD;
    // B-tile (4x16 f32): vgpr0 = row (l<16?0:2), vgpr1 = row (l<16?1:3)
    v2f bch0, bch1;
    bch0[0] = e[hi * 2 + 0];
    bch0[1] = e[hi * 2 + 1];
    bch1[0] = e[4 + hi * 2 + 0];
    bch1[1] = e[4 + hi * 2 + 1];

    float bv = -3.4e38f;
    int   bi = 0;
    for (int tile = 0; tile < 64; ++tile) {
      const float* a0p = sCbA + (tile * 64 + lane) * 2;
      const float* a1p = a0p + 64;
      v2f a0; a0[0] = a0p[0]; a0[1] = a0p[1];
      v2f a1; a1[0] = a1p[0]; a1[1] = a1p[1];
      v8f c = {};
      c = __builtin_amdgcn_wmma_f32_16x16x4_f32(false, a0, false, bch0, (short)0, c, false, false);
      c = __builtin_amdgcn_wmma_f32_16x16x4_f32(false, a1, false, bch1, (short)0, c, false, false);
#pragma unroll
      for (int v = 0; v < 8; ++v) {
        const int code = tile * 16 + hi * 8 + v;     // ascending within lane
        if (c[v] > bv) { bv = c[v]; bi = code; }     // strict > keeps lowest idx
      }
    }
    // merge the two half-rows (codes 0-7 vs 8-15 of each tile)
    const float ov = __shfl_xor(bv, 16, 32);
    const int   oi = __shfl_xor(bi, 16, 32);
    if (ov > bv || (ov == bv && oi < bi)) { bv = ov; bi = oi; }
    if (hi == 0) {
      sIdx[tl] = bi;
      o_idx[((size_t)b << 12) + tt0 + tl] = (float)bi;
    }
  }
  __syncthreads();

  // ---- Phase 4: loss partial: sum (z_e - codebook[idx])^2 over this block ----
  const int myi = sIdx[tid];
  const float4* cb4 = (const float4*)(codebook + myi * CD);
  const float4 q0 = cb4[0], q1 = cb4[1];
  const float qv[CD] = {q0.x, q0.y, q0.z, q0.w, q1.x, q1.y, q1.z, q1.w};
  float ls = 0.f;
#pragma unroll
  for (int o = 0; o < CD; ++o) {
    const float d2 = acc[o] - qv[o];
    ls = fmaf(d2, d2, ls);
  }
  sRed[tid] = ls;
  __syncthreads();
  for (int w = 128; w > 0; w >>= 1) {
    if (tid < w) sRed[tid] += sRed[tid + w];
    __syncthreads();
  }
  if (tid == 0) partial[blk] = sRed[0];
}

// ---------------------------------------------------------------------------
// Out projection: z_q_out[b,:,t] = W_out @ codebook[idx[b,t]] + b_out
// ---------------------------------------------------------------------------
__global__ void __launch_bounds__(256) k_out(const float* __restrict__ codebook,
                                             const float* __restrict__ b_out,
                                             const float* __restrict__ Wo,
                                             const float* __restrict__ o_idx,
                                             float* __restrict__ o_zq) {
  __shared__ float sWo[DIM * CD];
  __shared__ float sB[DIM];
  const int tid = threadIdx.x;
  for (int i = tid; i < DIM * CD; i += 256) sWo[i] = Wo[i];
  for (int i = tid; i < DIM; i += 256) sB[i] = b_out[i];
  __syncthreads();

  const int blk = blockIdx.x;
  const int b   = blk >> 4;
  const int t   = ((blk & 15) << 8) + tid;

  const int idx = (int)o_idx[((size_t)b << 12) + t];
  const float4* cb4 = (const float4*)(codebook + idx * CD);
  const float4 q0 = cb4[0], q1 = cb4[1];

  float* dst = o_zq + ((size_t)b << 22) + t;
#pragma unroll 4
  for (int d = 0; d < DIM; ++d) {
    const float4* w4 = (const float4*)(sWo + d * CD);
    const float4 w0 = w4[0], w1 = w4[1];
    float r = sB[d];
    r = fmaf(w0.x, q0.x, r);
    r = fmaf(w0.y, q0.y, r);
    r = fmaf(w0.z, q0.z, r);
    r = fmaf(w0.w, q0.w, r);
    r = fmaf(w1.x, q1.x, r);
    r = fmaf(w1.y, q1.y, r);
    r = fmaf(w1.z, q1.z, r);
    r = fmaf(w1.w, q1.w, r);
    dst[(size_t)d << 12] = r;
  }
}

// ---------------------------------------------------------------------------
// Final deterministic loss reduction: mean over CD*T = 32768 elements.
// commitment_loss == codebook_loss in forward.
// ---------------------------------------------------------------------------
__global__ void k_loss(const float* __restrict__ partial,
                       float* __restrict__ o_commit,
                       float* __restrict__ o_cb) {
  const int b = threadIdx.x;
  if (b < BATCH) {
    float s = 0.f;
    for (int j = 0; j < 16; ++j) s += partial[b * 16 + j];
    const float v = s / (float)(CD * TLEN);
    o_commit[b] = v;
    o_cb[b]     = v;
  }
}

// ---------------------------------------------------------------------------
extern "C" void kernel_launch(void* const* d_in, const int* in_sizes, int n_in,
                              void* d_out, int out_size, void* d_ws, size_t ws_size,
                              hipStream_t stream) {
  const float* z     = (const float*)d_in[0];
  const float* V_in  = (const float*)d_in[1];
  const float* g_in  = (const float*)d_in[2];
  const float* b_in  = (const float*)d_in[3];
  const float* cb    = (const float*)d_in[4];
  const float* V_out = (const float*)d_in[5];
  const float* g_out = (const float*)d_in[6];
  const float* b_out = (const float*)d_in[7];

  float* ws      = (float*)d_ws;
  float* Wt      = ws;            // 8192 floats: W_in^T [d][8]
  float* cbA     = ws + 8192;     // 8192 floats: WMMA-A packed normalized codebook
  float* Wo      = ws + 16384;    // 8192 floats: W_out [d][8]
  float* partial = ws + 24576;    // 256 floats: per-block loss partials

  float* out     = (float*)d_out;
  float* o_zq    = out;                                    // [16,1024,4096]
  float* o_comm  = out + (size_t)BATCH * DIM * TLEN;       // [16]
  float* o_cbl   = o_comm + BATCH;                         // [16]
  float* o_idx   = o_cbl + BATCH;                          // [16,4096] (as float)
  float* o_ze    = o_idx + (size_t)BATCH * TLEN;           // [16,8,4096]

  k_prep_win<<<1, 256, 0, stream>>>(V_in, g_in, Wt);
  k_prep_rows<<<8, 256, 0, stream>>>(V_out, g_out, cb, Wo, cbA);
  k_main<<<BATCH * (TLEN / TPB), TPB, 0, stream>>>(z, b_in, cb, Wt, cbA,
                                                   o_idx, o_ze, partial);
  k_out<<<BATCH * (TLEN / TPB), TPB, 0, stream>>>(cb, b_out, Wo, o_idx, o_zq);
  k_loss<<<1, 32, 0, stream>>>(partial, o_comm, o_cbl);
}